// QDNN_8899172237753
// MI455X (gfx1250) — compile-verified
//
#include <hip/hip_runtime.h>
#include <math.h>

typedef __attribute__((ext_vector_type(2))) float v2f;
typedef __attribute__((ext_vector_type(8))) float v8f;

#define B_   64
#define L_   512
#define D_   50
#define M_   128
#define KP   64          // K padded to 64 (zeros beyond 50)
#define LSPLIT 8         // sequence split across blocks
#define LCHUNK (L_ / LSPLIT)

// ---------------- kernel 0: zero the output (we atomically accumulate) --------
__global__ void zero_kernel(float* __restrict__ p, int n) {
  int i = blockIdx.x * blockDim.x + threadIdx.x;
  if (i < n) p[i] = 0.0f;
}

// ---------------- kernel 1: w = softmax over sequence of gathered mix logits --
__global__ void softmax_kernel(const int* __restrict__ x,
                               const float* __restrict__ mix_table,
                               float* __restrict__ w) {
  __shared__ float red[L_];
  const int b = blockIdx.x, t = threadIdx.x;
  const float v = mix_table[x[b * L_ + t]];
  red[t] = v;
  __syncthreads();
  for (int s = L_ / 2; s > 0; s >>= 1) {
    if (t < s) red[t] = fmaxf(red[t], red[t + s]);
    __syncthreads();
  }
  const float mx = red[0];
  __syncthreads();
  const float e = expf(v - mx);
  red[t] = e;
  __syncthreads();
  for (int s = L_ / 2; s > 0; s >>= 1) {
    if (t < s) red[t] += red[t + s];
    __syncthreads();
  }
  w[b * L_ + t] = e / red[0];
}

// ---------------- kernel 2: row-normalize measurement vectors, pad K to 64 ----
__global__ void knorm_kernel(const float* __restrict__ kr_in,
                             const float* __restrict__ ki_in,
                             float* __restrict__ krw,
                             float* __restrict__ kiw) {
  const int m = blockIdx.x * blockDim.x + threadIdx.x;
  if (m >= M_) return;
  float ss = 0.0f;
  for (int d = 0; d < D_; ++d) {
    const float a = kr_in[m * D_ + d];
    const float c = ki_in[m * D_ + d];
    ss += a * a + c * c;
  }
  const float rn = 1.0f / sqrtf(ss);
  for (int d = 0; d < KP; ++d) {
    krw[m * KP + d] = (d < D_) ? kr_in[m * D_ + d] * rn : 0.0f;
    kiw[m * KP + d] = (d < D_) ? ki_in[m * D_ + d] * rn : 0.0f;
  }
}

// ---------------- kernel 3: p[b,m] += sum_l w_l * |m_m^H v_l|^2 via f32 WMMA --
// grid = (B_, LSPLIT), block = 256 (8 waves); wave w owns m in [16w, 16w+16)
__global__ void __launch_bounds__(256)
qproj_kernel(const int* __restrict__ x,
             const float* __restrict__ amp_table,
             const float* __restrict__ pha_table,
             const float* __restrict__ w,
             const float* __restrict__ krw,
             const float* __restrict__ kiw,
             float* __restrict__ out) {
  extern __shared__ float smem[];
  float* krs  = smem;                 // [M_][KP]  normalized kernel_real
  float* kis  = krs + M_ * KP;        // [M_][KP]  normalized kernel_imag
  float* vrt  = kis + M_ * KP;        // [16][KP]  tile of vr = amp*cos(pha)
  float* vit  = vrt + 16 * KP;        // [16][KP]  tile of vi = amp*sin(pha)
  float* wt   = vit + 16 * KP;        // [16]      tile of softmax weights
  float* pacc = wt + 16;              // [M_]      per-block partial of p[b,:]

  const int b    = blockIdx.x;
  const int lt0  = blockIdx.y * LCHUNK;
  const int tid  = threadIdx.x;
  const int wave = tid >> 5;
  const int lane = tid & 31;

  // stage normalized measurement vectors once
  for (int i = tid; i < M_ * KP; i += 256) {
    krs[i] = krw[i];
    kis[i] = kiw[i];
  }
  if (tid < M_) pacc[tid] = 0.0f;
  __syncthreads();

  const int mbase = wave * 16;        // N-tile base (m index)
  const int nrow  = lane & 15;        // A row (token) / B col (m) for fragment fetch
  const int khalf = (lane >> 4) * 2;  // ISA layout: lanes 16-31 hold K+2, K+3
  const int rbase = (lane >> 4) * 8;  // C layout: lanes 16-31 hold rows M+8

  for (int lt = lt0; lt < lt0 + LCHUNK; lt += 16) {
    // stage 16 tokens: vr, vi over padded K (16*64 = 1024 entries, 4 per thread)
    for (int i = tid; i < 16 * KP; i += 256) {
      const int tok = lt + (i >> 6);
      const int d   = i & (KP - 1);
      float vr = 0.0f, vi = 0.0f;
      if (d < D_) {
        const int   idx = x[b * L_ + tok];
        const float a   = amp_table[idx * D_ + d];
        const float ph  = pha_table[idx * D_ + d];
        float s, c;
        sincosf(ph, &s, &c);
        vr = a * c;
        vi = a * s;
      }
      vrt[i] = vr;
      vit[i] = vi;
    }
    if (tid < 16) wt[tid] = w[b * L_ + lt + tid];
    __syncthreads();

    // GEMM [16 tokens x 64] x [64 x 16 m] with V_WMMA_F32_16X16X4_F32
    v8f cRR = {}; // kr . vr
    v8f cII = {}; // ki . vi
    v8f cIR = {}; // kr . vi
    v8f cRI = {}; // ki . vr
    #pragma unroll
    for (int ks = 0; ks < KP; ks += 4) {
      const int ka = ks + khalf;
      v2f aR = { vrt[nrow * KP + ka], vrt[nrow * KP + ka + 1] };
      v2f aI = { vit[nrow * KP + ka], vit[nrow * KP + ka + 1] };
      v2f bR = { krs[(mbase + nrow) * KP + ka], krs[(mbase + nrow) * KP + ka + 1] };
      v2f bI = { kis[(mbase + nrow) * KP + ka], kis[(mbase + nrow) * KP + ka + 1] };
      cRR = __builtin_amdgcn_wmma_f32_16x16x4_f32(false, aR, false, bR, (short)0, cRR, false, false);
      cII = __builtin_amdgcn_wmma_f32_16x16x4_f32(false, aI, false, bI, (short)0, cII, false, false);
      cIR = __builtin_amdgcn_wmma_f32_16x16x4_f32(false, aI, false, bR, (short)0, cIR, false, false);
      cRI = __builtin_amdgcn_wmma_f32_16x16x4_f32(false, aR, false, bI, (short)0, cRI, false, false);
    }

    // Ar = kr.vr + ki.vi ; Ai = kr.vi - ki.vr ; fold w_l * (Ar^2 + Ai^2)
    float partial = 0.0f;
    #pragma unroll
    for (int r = 0; r < 8; ++r) {
      const float ar = cRR[r] + cII[r];
      const float ai = cIR[r] - cRI[r];
      partial = fmaf(wt[rbase + r], fmaf(ar, ar, ai * ai), partial);
    }
    // combine the two 8-row halves of the C matrix (lane n <-> lane n+16, same m)
    partial += __shfl_xor(partial, 16, 32);
    if (lane < 16) pacc[mbase + lane] += partial;
    __syncthreads();
  }

  if (tid < M_) {
    // fast HW float atomic (global_atomic_add_f32) to merge sequence splits
    unsafeAtomicAdd(&out[b * M_ + tid], pacc[tid]);
  }
}

extern "C" void kernel_launch(void* const* d_in, const int* in_sizes, int n_in,
                              void* d_out, int out_size, void* d_ws, size_t ws_size,
                              hipStream_t stream) {
  const int*   x   = (const int*)d_in[0];
  const float* amp = (const float*)d_in[1];
  const float* pha = (const float*)d_in[2];
  const float* mix = (const float*)d_in[3];
  const float* kr  = (const float*)d_in[4];
  const float* ki  = (const float*)d_in[5];
  float*       out = (float*)d_out;

  // workspace layout: w[B*L] | krw[M*KP] | kiw[M*KP]
  float* w   = (float*)d_ws;
  float* krw = w + B_ * L_;
  float* kiw = krw + M_ * KP;

  zero_kernel<<<(B_ * M_ + 255) / 256, 256, 0, stream>>>(out, B_ * M_);
  softmax_kernel<<<B_, L_, 0, stream>>>(x, mix, w);
  knorm_kernel<<<1, M_, 0, stream>>>(kr, ki, krw, kiw);

  const size_t lds_bytes =
      (size_t)(M_ * KP * 2 + 16 * KP * 2 + 16 + M_) * sizeof(float);
  dim3 grid(B_, LSPLIT);
  qproj_kernel<<<grid, 256, lds_bytes, stream>>>(x, amp, pha, w, krw, kiw, out);
}